// MoDLayer_29643864277389
// MI455X (gfx1250) — compile-verified
//
#include <hip/hip_runtime.h>

// ---------------- CDNA5 (gfx1250) fused Mixture-of-Depths layer ----------------
// fp32 WMMA (V_WMMA_F32_16X16X4_F32) for full fp32 fidelity vs the reference.
// MT=32 token tile: each W1/W2 fragment loaded from L2 feeds 2 (phase1) or
// 2 (phase2) WMMAs -> 16 FLOP per L2 byte on the dominant weight stream.

typedef float v2f __attribute__((ext_vector_type(2)));
typedef float v8f __attribute__((ext_vector_type(8)));

#define DIM   1024
#define HID   4096
#define NTOK  16384          // 4 * 4096 tokens
#define MT    32             // token tile (rows per workgroup)
#define HC    128            // hidden chunk held in LDS
#define NCHUNK (HID / HC)    // 32
#define XS_STRIDE (DIM + 4)  // pad -> lane-m A reads land on distinct banks
#define HS_STRIDE (HC + 4)

// ---------------------------------------------------------------- init counter
__global__ void mod_init(int* __restrict__ count) {
    if (threadIdx.x == 0) *count = 0;
}

// ------------------------------------------------------------------ router
// one wave32 per token: logit = dot(x_tok, W_r) + b_r; active iff logit > 0
__global__ __launch_bounds__(256) void mod_router(
    const float* __restrict__ x, const float* __restrict__ Wr,
    const float* __restrict__ br, int* __restrict__ count,
    int* __restrict__ active)
{
    const int wave  = threadIdx.x >> 5;
    const int lane  = threadIdx.x & 31;
    const int token = blockIdx.x * 8 + wave;
    const float* xr = x + (size_t)token * DIM;
    float s = 0.f;
    #pragma unroll 8
    for (int i = lane; i < DIM; i += 32) s += xr[i] * Wr[i];
    #pragma unroll
    for (int off = 16; off; off >>= 1) s += __shfl_xor(s, off, 32);
    if (lane == 0) {
        const float logit = s + br[0];
        if (logit > 0.f) {                 // sigmoid(logit) > 0.5
            const int p = atomicAdd(count, 1);
            active[p] = token;
        }
    }
}

// ------------------------------------------------------------------ frac
__global__ void mod_frac(const int* __restrict__ count, float* __restrict__ outf) {
    if (threadIdx.x == 0) *outf = (float)(*count) * (1.0f / (float)NTOK);
}

// ------------------------------------------------------------------ fused MLP
// 32 routed tokens per workgroup, 8 waves. Hidden dim processed in 128-wide
// chunks so only a 32x128 slice of the activation sits in LDS at a time.
__global__ __launch_bounds__(256) void mod_mlp(
    const float* __restrict__ x,
    const float* __restrict__ W1, const float* __restrict__ b1,
    const float* __restrict__ W2, const float* __restrict__ b2,
    const int* __restrict__ count, const int* __restrict__ active,
    float* __restrict__ out)
{
    __shared__ __align__(16) float xs[MT * XS_STRIDE]; // 32 x 1024 x-tile (~128.5 KB)
    __shared__ __align__(16) float hs[MT * HS_STRIDE]; // 32 x 128 hidden chunk (~16.5 KB)
    __shared__ int tok[MT];

    const int nAct   = *count;            // uniform across block
    const int tstart = blockIdx.x * MT;
    if (tstart >= nAct) return;           // uniform branch

    const int tid   = threadIdx.x;
    const int wave  = tid >> 5;
    const int lane  = tid & 31;
    const int m     = lane & 15;          // A row / B col within a 16-tile
    const int khalf = lane >> 4;          // 0: K={0,1}, 1: K={2,3}

    if (tid < MT) {
        const int slot = tstart + tid;
        tok[tid] = (slot < nAct) ? active[slot] : -1;
    }
    __syncthreads();

    // cooperative float4 load of the x-tile (zeros for invalid slots)
    for (int i = tid; i < MT * (DIM / 4); i += 256) {
        const int r  = i >> 8;            // DIM/4 == 256
        const int c4 = (i & 255) * 4;
        float4 v = make_float4(0.f, 0.f, 0.f, 0.f);
        if (tok[r] >= 0) v = *(const float4*)(x + (size_t)tok[r] * DIM + c4);
        *(float4*)(&xs[r * XS_STRIDE + c4]) = v;
    }
    __syncthreads();

    // output accumulators: this wave owns cols [wave*128, wave*128+128)
    //   oacc[t]   : rows  0..15, cols t*16..t*16+15
    //   oacc[8+t] : rows 16..31, same cols
    v8f oacc[16];
    #pragma unroll
    for (int t = 0; t < 16; ++t) oacc[t] = (v8f)0.0f;

    for (int ch = 0; ch < NCHUNK; ++ch) {
        // warm next chunk's weight panels in GL2 (global_prefetch_b8)
        if (ch + 1 < NCHUNK) {
            __builtin_prefetch(W1 + (size_t)(ch + 1) * HC + m * 64, 0, 1);
            __builtin_prefetch(W2 + (size_t)(ch + 1) * HC * DIM + m * 64, 0, 1);
        }

        // ---- phase 1: hs = relu(xs @ W1[:, ch*HC : +HC] + b1) ----
        // wave owns 16 chunk-cols; each W1 fragment feeds both row-halves.
        const int n0   = wave * 16;               // chunk-local col base
        const int gcol = ch * HC + n0 + m;        // global hidden col
        v8f accA = (v8f)0.0f, accB = (v8f)0.0f;
        #pragma unroll 4
        for (int k = 0; k < DIM; k += 4) {
            const int kk = k + 2 * khalf;
            v2f a0; a0.x = xs[m * XS_STRIDE + kk];
                    a0.y = xs[m * XS_STRIDE + kk + 1];
            v2f a1; a1.x = xs[(16 + m) * XS_STRIDE + kk];
                    a1.y = xs[(16 + m) * XS_STRIDE + kk + 1];
            v2f b;  b.x  = W1[(size_t)kk * HID + gcol];
                    b.y  = W1[(size_t)(kk + 1) * HID + gcol];
            accA = __builtin_amdgcn_wmma_f32_16x16x4_f32(false, a0, false, b,
                                                         (short)0, accA, false, false);
            accB = __builtin_amdgcn_wmma_f32_16x16x4_f32(false, a1, false, b,
                                                         (short)0, accB, false, false);
        }
        const float bb = b1[gcol];
        #pragma unroll
        for (int v = 0; v < 8; ++v) {
            const int row = v + 8 * khalf;        // C layout: lanes16-31 -> M+8
            float h0 = accA[v] + bb; h0 = h0 > 0.f ? h0 : 0.f;
            float h1 = accB[v] + bb; h1 = h1 > 0.f ? h1 : 0.f;
            hs[row * HS_STRIDE + n0 + m]        = h0;   // rows  0..15
            hs[(16 + row) * HS_STRIDE + n0 + m] = h1;   // rows 16..31
        }
        __syncthreads();

        // ---- phase 2: oacc += hs @ W2[ch*HC : +HC, :] ----
        // each W2 fragment feeds 2 WMMAs (both row-halves of the token tile).
        #pragma unroll 2
        for (int k = 0; k < HC; k += 4) {
            const int kk = k + 2 * khalf;
            v2f a0; a0.x = hs[m * HS_STRIDE + kk];
                    a0.y = hs[m * HS_STRIDE + kk + 1];
            v2f a1; a1.x = hs[(16 + m) * HS_STRIDE + kk];
                    a1.y = hs[(16 + m) * HS_STRIDE + kk + 1];
            const int kg = ch * HC + kk;
            #pragma unroll
            for (int t = 0; t < 8; ++t) {
                const int colg = wave * 128 + t * 16 + m;
                v2f b; b.x = W2[(size_t)kg * DIM + colg];
                       b.y = W2[(size_t)(kg + 1) * DIM + colg];
                oacc[t]     = __builtin_amdgcn_wmma_f32_16x16x4_f32(false, a0, false, b,
                                                    (short)0, oacc[t],     false, false);
                oacc[8 + t] = __builtin_amdgcn_wmma_f32_16x16x4_f32(false, a1, false, b,
                                                    (short)0, oacc[8 + t], false, false);
            }
        }
        __syncthreads();
    }

    // ---- scatter: out[tok] = oacc + b2 (only valid slots) ----
    #pragma unroll
    for (int t = 0; t < 8; ++t) {
        const int colg = wave * 128 + t * 16 + m;
        const float bias = b2[colg];
        #pragma unroll
        for (int v = 0; v < 8; ++v) {
            const int row = v + 8 * khalf;
            if (tstart + row < nAct) {
                out[(size_t)tok[row] * DIM + colg] = oacc[t][v] + bias;
            }
            if (tstart + 16 + row < nAct) {
                out[(size_t)tok[16 + row] * DIM + colg] = oacc[8 + t][v] + bias;
            }
        }
    }
}

// ------------------------------------------------------------------ launch
extern "C" void kernel_launch(void* const* d_in, const int* in_sizes, int n_in,
                              void* d_out, int out_size, void* d_ws, size_t ws_size,
                              hipStream_t stream) {
    const float* x  = (const float*)d_in[0];
    const float* Wr = (const float*)d_in[1];
    const float* br = (const float*)d_in[2];
    const float* W1 = (const float*)d_in[3];
    const float* b1 = (const float*)d_in[4];
    const float* W2 = (const float*)d_in[5];
    const float* b2 = (const float*)d_in[6];
    float* out      = (float*)d_out;

    int* count  = (int*)d_ws;       // [1]
    int* active = count + 1;        // [NTOK]

    mod_init<<<1, 32, 0, stream>>>(count);
    mod_router<<<NTOK / 8, 256, 0, stream>>>(x, Wr, br, count, active);

    // pass-through default: out = x (active rows get overwritten by mod_mlp)
    hipMemcpyAsync(out, x, (size_t)NTOK * DIM * sizeof(float),
                   hipMemcpyDeviceToDevice, stream);

    mod_frac<<<1, 32, 0, stream>>>(count, out + (size_t)NTOK * DIM);

    mod_mlp<<<NTOK / MT, 256, 0, stream>>>(x, W1, b1, W2, b2, count, active, out);
}